// OnlineKNN_31138512896770
// MI455X (gfx1250) — compile-verified
//
#include <hip/hip_runtime.h>
#include <cstdint>

#define NROWS   1024
#define EDIM    512
#define QCOLS   65536
#define NKNN    200
#define NCLS    1000
#define LDB     516   // padded LDS column stride (floats): 516*4 is 16B-aligned,
                      // 516 % 64 == 4 banks/col -> conflict-free ds_load_b64 frags
#define TKT     512   // top-k block threads (16 waves)

typedef __attribute__((ext_vector_type(2))) float v2f;
typedef __attribute__((ext_vector_type(4))) float v4f;
typedef __attribute__((ext_vector_type(8))) float v8f;

// ---------------------------------------------------------------------------
// Kernel 1: sim[b][q] = sum_e feat[b][e] * qf[q][e]  (q = k*1024+n; qf row
// (k,n) is 512 contiguous floats, i.e. B is column-contiguous).
// f32 WMMA 16x16x4; B tile (16 cols x 512) staged in LDS via b128 loads and
// reused by 8 waves x 8 M-tiles. Epilogue stores order-preserving uint keys
// (monotone float->uint map) so the top-k kernel consumes them directly.
// ---------------------------------------------------------------------------
__global__ void __launch_bounds__(256) knn_gemm_wmma_f32(
    const float* __restrict__ feat,      // [1024,512]
    const float* __restrict__ qf,        // [65536,512]
    unsigned int* __restrict__ simkey)   // [1024,65536] order-preserving keys
{
    __shared__ float bs[16 * LDB];       // 33 KB B tile, padded

    const int q0 = blockIdx.x * 16;
    const int t  = threadIdx.x;

    // Cooperative b128 staging: 16 cols x 128 float4s, coalesced along e.
    for (int i = t; i < 16 * (EDIM / 4); i += 256) {
        const int col = i >> 7;              // i / 128
        const int e4  = i & 127;
        *(v4f*)&bs[col * LDB + 4 * e4] =
            *(const v4f*)&qf[(size_t)(q0 + col) * EDIM + 4 * e4];
    }
    __syncthreads();

    const int wave = t >> 5;          // wave32: 8 waves per block
    const int lane = t & 31;
    const int lh   = lane >> 4;       // lane-half: selects K+2 (A/B), M+8 (C)
    const int ll   = lane & 15;

    // Each wave computes 8 of the 64 M-tiles for this column tile.
    for (int mt = wave; mt < 64; mt += 8) {
        const int m0 = mt * 16;
        v8f acc = {};
        const float* arow = feat + (size_t)(m0 + ll) * EDIM + 2 * lh;
        const float* bcol = &bs[ll * LDB + 2 * lh];
        #pragma unroll 8
        for (int k0 = 0; k0 < EDIM; k0 += 4) {
            v2f a = *(const v2f*)(arow + k0);   // A[M=ll][k0+2*lh .. +1]
            v2f b = *(const v2f*)(bcol + k0);   // B[k0+2*lh .. +1][N=ll] (LDS)
            acc = __builtin_amdgcn_wmma_f32_16x16x4_f32(
                false, a, false, b, (short)0, acc, false, false);
        }
        // C layout: lane ll = column, VGPR v = row (v + 8*lh).
        // Fused epilogue: float -> ascending-order uint key.
        unsigned int* cbase =
            simkey + (size_t)(m0 + 8 * lh) * QCOLS + (q0 + ll);
        #pragma unroll
        for (int v = 0; v < 8; ++v) {
            const unsigned int b = __float_as_uint(acc[v]);
            cbase[(size_t)v * QCOLS] =
                b ^ ((b >> 31) ? 0xFFFFFFFFu : 0x80000000u);
        }
    }
}

// ---------------------------------------------------------------------------
// Kernel 2: per-row exact top-200 via 4-pass MSB-first radix select over the
// precomputed keys, cached in 256 KB of LDS (CDNA5: 320 KB/WGP). 512 threads
// (16 waves) per block to hide latency at 1 block/WGP occupancy. Then
// exp-weighted class voting (ds_add_f32), index-stable argmax, accuracy.
// ---------------------------------------------------------------------------
__global__ void __launch_bounds__(TKT) knn_topk_vote(
    const unsigned int* __restrict__ simkey,  // [1024,65536]
    const int*          __restrict__ labels,  // [1024]
    const int*          __restrict__ qlabels, // [65536]
    float*              __restrict__ out)     // [1] accuracy
{
    extern __shared__ unsigned int keys[];    // 65536 * 4B = 256 KB (dynamic)
    __shared__ unsigned int hist[256];
    __shared__ float scores[NCLS];
    __shared__ float rv[TKT];
    __shared__ int   ri[TKT];
    __shared__ unsigned int s_prefix;
    __shared__ int          s_rem;
    __shared__ unsigned int s_eq;

    const int row = blockIdx.x;
    const int t   = threadIdx.x;
    const unsigned int* srow = simkey + (size_t)row * QCOLS;

    // Cache the whole row of keys in LDS (single 256 KB streaming read).
    for (int i = t; i < QCOLS; i += TKT)
        keys[i] = srow[i];
    if (t == 0) { s_prefix = 0u; s_rem = NKNN; s_eq = 0u; }
    for (int i = t; i < NCLS; i += TKT) scores[i] = 0.0f;
    __syncthreads();

    // MSB-first radix select: key value of the 200th largest element.
    for (int shift = 24; shift >= 0; shift -= 8) {
        if (t < 256) hist[t] = 0u;
        __syncthreads();
        const unsigned int pfx   = s_prefix;
        const unsigned int hmask = (shift == 24) ? 0u : (0xFFFFFFFFu << (shift + 8));
        for (int i = t; i < QCOLS; i += TKT) {
            const unsigned int k = keys[i];
            if ((k & hmask) == pfx)
                atomicAdd(&hist[(k >> shift) & 0xFFu], 1u);
        }
        __syncthreads();
        if (t == 0) {
            const int rem = s_rem;
            unsigned int cum = 0u;
            int bin = 255;
            for (; bin > 0; --bin) {
                cum += hist[bin];
                if ((int)cum >= rem) break;
            }
            if ((int)cum < rem) cum += hist[0];        // fell through to bin 0
            s_rem    = rem - (int)(cum - hist[bin]);   // remaining among equals
            s_prefix = pfx | ((unsigned int)bin << shift);
        }
        __syncthreads();
    }

    const unsigned int T  = s_prefix;   // key of the 200th largest
    const int eq_take     = s_rem;      // how many == T to include
    __syncthreads();

    // Voting: keys > T always taken; keys == T taken up to eq_take slots.
    for (int i = t; i < QCOLS; i += TKT) {
        const unsigned int k = keys[i];
        bool take = (k > T);
        if (!take && k == T) {
            const unsigned int slot = atomicAdd(&s_eq, 1u);
            take = (slot < (unsigned int)eq_take);
        }
        if (take) {
            const unsigned int b = (k >> 31) ? (k ^ 0x80000000u) : ~k;
            const float v = __uint_as_float(b);
            const float w = expf(v * (1.0f / 0.07f));
            atomicAdd(&scores[qlabels[i]], w);         // ds_add_f32
        }
    }
    __syncthreads();

    // Argmax over 1000 classes, lowest index wins ties (matches jnp.argmax).
    float bv = -1.0f; int bi = 0;
    for (int c = t; c < NCLS; c += TKT) {
        const float s = scores[c];
        if (s > bv) { bv = s; bi = c; }
    }
    rv[t] = bv; ri[t] = bi;
    __syncthreads();
    for (int off = TKT / 2; off > 0; off >>= 1) {
        if (t < off) {
            if (rv[t + off] > rv[t] ||
                (rv[t + off] == rv[t] && ri[t + off] < ri[t])) {
                rv[t] = rv[t + off]; ri[t] = ri[t + off];
            }
        }
        __syncthreads();
    }
    if (t == 0 && ri[0] == labels[row])
        atomicAdd(out, 1.0f / (float)NROWS);
}

__global__ void knn_zero_out(float* out) { out[0] = 0.0f; }

// ---------------------------------------------------------------------------
extern "C" void kernel_launch(void* const* d_in, const int* in_sizes, int n_in,
                              void* d_out, int out_size, void* d_ws, size_t ws_size,
                              hipStream_t stream) {
    const float* feat    = (const float*)d_in[0];   // [1024,512]
    const int*   labels  = (const int*)  d_in[1];   // [1024]
    const float* qf      = (const float*)d_in[2];   // [64,1024,512]
    const int*   qlabels = (const int*)  d_in[3];   // [64,1024] -> flat [65536]
    float*       out     = (float*)d_out;
    unsigned int* simkey = (unsigned int*)d_ws;     // 1024*65536*4 = 256 MB

    knn_zero_out<<<dim3(1), dim3(1), 0, stream>>>(out);

    knn_gemm_wmma_f32<<<dim3(QCOLS / 16), dim3(256), 0, stream>>>(feat, qf, simkey);

    knn_topk_vote<<<dim3(NROWS), dim3(TKT),
                    (size_t)QCOLS * sizeof(unsigned int), stream>>>(
        simkey, labels, qlabels, out);
}